// Encoder_21492016349714
// MI455X (gfx1250) — compile-verified
//
#include <hip/hip_runtime.h>
#include <hip/hip_bf16.h>

// ---------------------------------------------------------------------------
// Matching-pursuit encoder for MI455X (gfx1250, wave32).
// Core correlation = Toeplitz GEMM done with V_WMMA_F32_16X16X4_F32:
// each wave computes a 16(pos) x 16(atom) tile, K=256 in 64 chained WMMAs.
// ---------------------------------------------------------------------------

#define T_LEN    32768
#define KER      256
#define DSZ      16
#define NSTEPS   16
#define BATCH    32
#define CH       128
#define RSTRIDE  (T_LEN + 512)      // padded residual stride (>= T+KER-1 reach)
#define PPB      128                // positions per block (8 waves * 16)
#define NBLK     (T_LEN / PPB)      // 256 correlation blocks per batch

typedef __attribute__((ext_vector_type(2))) float v2f;
typedef __attribute__((ext_vector_type(8))) float v8f;

// ---- build zero-padded residual from x --------------------------------------
__global__ void k_init(const float* __restrict__ x, float* __restrict__ res) {
  int i = blockIdx.x * blockDim.x + threadIdx.x;
  if (i >= BATCH * RSTRIDE) return;
  int b = i / RSTRIDE;
  int t = i - b * RSTRIDE;
  res[i] = (t < T_LEN) ? x[b * T_LEN + t] : 0.0f;
}

// ---- full correlation via WMMA + fused per-block argmax ---------------------
__global__ __launch_bounds__(256) void k_corr_argmax(
    const float* __restrict__ res, const float* __restrict__ dict,
    float* __restrict__ pval, unsigned* __restrict__ pidx) {
  __shared__ float s_dict[DSZ * KER];          // 16 KB: d[a][k]
  __shared__ float s_res[PPB + KER + 8];       // residual window
  __shared__ float s_wval[8];
  __shared__ unsigned s_widx[8];

  const int blk   = blockIdx.x;
  const int b     = blk / NBLK;
  const int chunk = blk - b * NBLK;
  const int p0    = chunk * PPB;
  const int tid   = threadIdx.x;

  for (int i = tid; i < DSZ * KER; i += 256) s_dict[i] = dict[i];
  const float* rb = res + (size_t)b * RSTRIDE + p0;
  for (int i = tid; i < PPB + KER; i += 256) s_res[i] = rb[i];
  __syncthreads();

  const int wave = tid >> 5;
  const int lane = tid & 31;
  const int half = lane >> 4;     // K half-select per ISA A/B layout
  const int lm   = lane & 15;     // M for A-frag, N (atom) for B-frag
  const int pl0  = wave * 16;     // tile position base within block

  // A[m,k] = s_res[pl0 + m + k]; lanes 0-15 carry K={k0,k0+1}, 16-31 K={k0+2,k0+3}
  const float* ra = &s_res[pl0 + lm + 2 * half];
  // B[k,n] = d[n, k];  lane = n, same K half split
  const float* bb = &s_dict[lm * KER + 2 * half];

  v8f acc = {0.f, 0.f, 0.f, 0.f, 0.f, 0.f, 0.f, 0.f};
#pragma unroll
  for (int k0 = 0; k0 < KER; k0 += 4) {
    v2f af; af.x = ra[k0]; af.y = ra[k0 + 1];
    v2f bf; bf.x = bb[k0]; bf.y = bb[k0 + 1];
    acc = __builtin_amdgcn_wmma_f32_16x16x4_f32(
        /*neg_a=*/false, af, /*neg_b=*/false, bf,
        /*c_mod=*/(short)0, acc, /*reuse_a=*/false, /*reuse_b=*/false);
  }

  // C/D layout: VGPR r -> M = r (+8 for lanes 16-31), N = lane&15
  float best = -3.0e38f;
  unsigned bidx = 0;
  const int pbase = p0 + pl0 + 8 * half;
#pragma unroll
  for (int r = 0; r < 8; ++r) {
    float v = acc[r];
    unsigned idx = (unsigned)((pbase + r) * DSZ + lm);   // pos*16 + atom
    if (v > best) { best = v; bidx = idx; }
  }
  for (int off = 16; off > 0; off >>= 1) {
    float    ov = __shfl_xor(best, off, 32);
    unsigned oi = __shfl_xor(bidx, off, 32);
    if (ov > best) { best = ov; bidx = oi; }
  }
  if (lane == 0) { s_wval[wave] = best; s_widx[wave] = bidx; }
  __syncthreads();
  if (tid == 0) {
    float bv = s_wval[0]; unsigned bi = s_widx[0];
    for (int w = 1; w < 8; ++w)
      if (s_wval[w] > bv) { bv = s_wval[w]; bi = s_widx[w]; }
    pval[blk] = bv;
    pidx[blk] = bi;
  }
}

// ---- global argmax across partials, record event, update residual ----------
__global__ __launch_bounds__(256) void k_select_update(
    float* __restrict__ res, const float* __restrict__ dict,
    const float* __restrict__ pval, const unsigned* __restrict__ pidx,
    float* __restrict__ ev_amp, unsigned* __restrict__ ev_idx, int step) {
  __shared__ float sv[256];
  __shared__ unsigned si[256];
  const int b = blockIdx.x;
  const int t = threadIdx.x;
  sv[t] = pval[b * NBLK + t];
  si[t] = pidx[b * NBLK + t];
  __syncthreads();
  for (int off = 128; off > 0; off >>= 1) {
    if (t < off && sv[t + off] > sv[t]) { sv[t] = sv[t + off]; si[t] = si[t + off]; }
    __syncthreads();
  }
  const float amp = sv[0];
  const unsigned idx = si[0];
  if (t == 0) {
    ev_amp[b * NSTEPS + step] = amp;
    ev_idx[b * NSTEPS + step] = idx;
  }
  const unsigned pos  = idx / DSZ;
  const unsigned atom = idx % DSZ;
  // residual window update (padded region always in-bounds)
  res[(size_t)b * RSTRIDE + pos + t] -= amp * dict[atom * KER + t];
}

// ---- sort 16 events per batch by amplitude descending (== top_k order) -----
__global__ void k_sort(const float* __restrict__ ev_amp, const unsigned* __restrict__ ev_idx,
                       float* __restrict__ sv_amp, int* __restrict__ sv_pos,
                       int* __restrict__ sv_atom) {
  int b = threadIdx.x;
  if (b >= BATCH) return;
  float a[NSTEPS];
  unsigned ix[NSTEPS];
  for (int i = 0; i < NSTEPS; ++i) { a[i] = ev_amp[b * NSTEPS + i]; ix[i] = ev_idx[b * NSTEPS + i]; }
  for (int i = 1; i < NSTEPS; ++i) {           // insertion sort, descending amp
    float av = a[i]; unsigned iv = ix[i];
    int j = i - 1;
    while (j >= 0 && a[j] < av) { a[j + 1] = a[j]; ix[j + 1] = ix[j]; --j; }
    a[j + 1] = av; ix[j + 1] = iv;
  }
  for (int i = 0; i < NSTEPS; ++i) {
    sv_amp[b * NSTEPS + i]  = a[i];
    sv_pos[b * NSTEPS + i]  = (int)(ix[i] / DSZ);
    sv_atom[b * NSTEPS + i] = (int)(ix[i] % DSZ);
  }
}

// ---- fold the three 1x1 convs into FA/FB/FC/Wat ----------------------------
__global__ void k_fold(const float* __restrict__ paw, const float* __restrict__ pab,
                       const float* __restrict__ aw,  const float* __restrict__ ab,
                       const float* __restrict__ rw,  const float* __restrict__ rb,
                       float* __restrict__ FA, float* __restrict__ FB,
                       float* __restrict__ FC, float* __restrict__ Wat) {
  int o = threadIdx.x;
  if (o >= CH) return;
  const float* r0 = rw + o * 2 * CH;
  float fa = 0.f, fb = 0.f, fc = rb[o];
  for (int c = 0; c < CH; ++c) {
    float r1 = r0[c], r2 = r0[CH + c];
    fa += r1 * paw[2 * c];
    fb += r1 * paw[2 * c + 1];
    fc += r1 * pab[c] + r2 * ab[c];
  }
  FA[o] = fa; FB[o] = fb; FC[o] = fc;
  for (int a2 = 0; a2 < DSZ; ++a2) {
    float w = 0.f;
    for (int c = 0; c < CH; ++c) w += r0[CH + c] * aw[c * DSZ + a2];
    Wat[o * DSZ + a2] = w;
  }
}

// ---- final output: (B, CH, NSTEPS) -----------------------------------------
__global__ void k_out(const float* __restrict__ FA, const float* __restrict__ FB,
                      const float* __restrict__ FC, const float* __restrict__ Wat,
                      const float* __restrict__ sv_amp, const int* __restrict__ sv_pos,
                      const int* __restrict__ sv_atom, float* __restrict__ out) {
  int i = blockIdx.x * blockDim.x + threadIdx.x;
  if (i >= BATCH * CH * NSTEPS) return;
  int n = i & (NSTEPS - 1);
  int o = (i >> 4) & (CH - 1);
  int b = i >> 11;
  float posn = (float)sv_pos[b * NSTEPS + n] * (1.0f / (float)T_LEN);
  float amp  = sv_amp[b * NSTEPS + n];
  int atom   = sv_atom[b * NSTEPS + n];
  out[i] = FC[o] + FA[o] * posn + FB[o] * amp + Wat[o * DSZ + atom];
}

extern "C" void kernel_launch(void* const* d_in, const int* in_sizes, int n_in,
                              void* d_out, int out_size, void* d_ws, size_t ws_size,
                              hipStream_t stream) {
  const float* x   = (const float*)d_in[0];   // (B,1,T)
  const float* d   = (const float*)d_in[1];   // (16,256)
  const float* paw = (const float*)d_in[2];   // (128,2)
  const float* pab = (const float*)d_in[3];   // (128,)
  const float* aw  = (const float*)d_in[4];   // (128,16)
  const float* ab  = (const float*)d_in[5];   // (128,)
  const float* rw  = (const float*)d_in[6];   // (128,256)
  const float* rb  = (const float*)d_in[7];   // (128,)
  float* out = (float*)d_out;

  // workspace carve-up (~4.35 MB)
  float* res = (float*)d_ws;                       // BATCH*RSTRIDE
  float* pval = res + (size_t)BATCH * RSTRIDE;     // BATCH*NBLK = 8192
  unsigned* pidx = (unsigned*)(pval + BATCH * NBLK);
  float* ev_amp = (float*)(pidx + BATCH * NBLK);   // 512
  unsigned* ev_idx = (unsigned*)(ev_amp + BATCH * NSTEPS);
  float* sv_amp = (float*)(ev_idx + BATCH * NSTEPS);
  int* sv_pos  = (int*)(sv_amp + BATCH * NSTEPS);
  int* sv_atom = sv_pos + BATCH * NSTEPS;
  float* FA  = (float*)(sv_atom + BATCH * NSTEPS);
  float* FB  = FA + CH;
  float* FC  = FB + CH;
  float* Wat = FC + CH;                            // CH*DSZ

  {
    int n = BATCH * RSTRIDE;
    k_init<<<(n + 255) / 256, 256, 0, stream>>>(x, res);
  }
  for (int step = 0; step < NSTEPS; ++step) {
    k_corr_argmax<<<BATCH * NBLK, 256, 0, stream>>>(res, d, pval, pidx);
    k_select_update<<<BATCH, 256, 0, stream>>>(res, d, pval, pidx, ev_amp, ev_idx, step);
  }
  k_sort<<<1, 32, 0, stream>>>(ev_amp, ev_idx, sv_amp, sv_pos, sv_atom);
  k_fold<<<1, CH, 0, stream>>>(paw, pab, aw, ab, rw, rb, FA, FB, FC, Wat);
  {
    int n = BATCH * CH * NSTEPS;
    k_out<<<(n + 255) / 256, 256, 0, stream>>>(FA, FB, FC, Wat, sv_amp, sv_pos, sv_atom, out);
  }
}